// VN_Cross_Gating_35880156790883
// MI455X (gfx1250) — compile-verified
//
#include <hip/hip_runtime.h>

// VN_Cross_Gating for MI455X (gfx1250, wave32).
// B=2, C=64, U=3, N=M=1024. All math fp32 via V_WMMA_F32_16X16X4_F32.

typedef __attribute__((ext_vector_type(2))) float v2f;
typedef __attribute__((ext_vector_type(8))) float v8f;

#define B_DIM 2
#define C_DIM 64
#define U_DIM 3
#define N_DIM 1024

// ---------------------------------------------------------------------------
// proj: Out[b,d,u,i] = sum_c W[d,c] * X[b,c,u,i]
// One wave computes one 16(d) x 16(i) tile, K-loop over c in steps of 4.
// Grid: 6 (b,u) pairs * 256 tiles / 8 waves-per-block = 192 blocks of 256.
// ---------------------------------------------------------------------------
__global__ __launch_bounds__(256) void proj_wmma(const float* __restrict__ W,
                                                 const float* __restrict__ X,
                                                 float* __restrict__ Q) {
  const int lane    = threadIdx.x & 31;
  const int wave    = threadIdx.x >> 5;
  const int bu      = blockIdx.x >> 5;                 // 0..5
  const int tile    = ((blockIdx.x & 31) << 3) + wave; // 0..255
  const int b       = bu / U_DIM;
  const int u       = bu % U_DIM;
  const int d0      = (tile >> 6) << 4;                // 4 d-tiles
  const int i0      = (tile & 63) << 4;                // 64 i-tiles
  const int halfsel = lane >> 4;                       // 0: K=0,1  1: K=2,3
  const int lrow    = lane & 15;

  v8f acc = {};
  for (int c0 = 0; c0 < C_DIM; c0 += 4) {
    const int k0 = c0 + (halfsel << 1);
    v2f a, bm;
    // A tile (W): 32-bit A 16x4 layout — lane holds row d0+lrow, K = k0, k0+1
    a.x = W[(d0 + lrow) * C_DIM + k0];
    a.y = W[(d0 + lrow) * C_DIM + k0 + 1];
    // B tile (X): rows K=k0,k0+1 striped across lanes at column i0+lrow
    const float* xp = X + (((size_t)(b * C_DIM + k0) * U_DIM + u) * N_DIM) + i0 + lrow;
    bm.x = xp[0];
    bm.y = xp[(size_t)U_DIM * N_DIM];                  // c -> c+1 stride
    acc = __builtin_amdgcn_wmma_f32_16x16x4_f32(false, a, false, bm,
                                                (short)0, acc, false, false);
  }
#pragma unroll
  for (int r = 0; r < 8; ++r) {
    const int row = d0 + r + (halfsel << 3);
    Q[(((size_t)(b * C_DIM + row) * U_DIM + u) * N_DIM) + i0 + lrow] = acc[r];
  }
}

// ---------------------------------------------------------------------------
// relmax: rel[b,c,i] = 0.125 * max_j sum_u q[b,c,u,i]*k[b,c,u,j]
// K-dim (u) = 3, padded to 4 with zeros -> ONE wmma per 16x16 logits tile.
// LDS holds the B operand PRE-PACKED per half-wave: kPair[0][j] = {k0,k1}
// (lanes 0-15), kPair[1][j] = {k2, 0} (lanes 16-31) -> inner loop is a
// single ds_load_b64 + wmma + 8 co-executable v_max per tile.
// Grid: (128 bc, 8 i-groups); each of the 8 waves owns one 16-row i-strip.
// ---------------------------------------------------------------------------
__global__ __launch_bounds__(256) void relmax_wmma(const float* __restrict__ Q,
                                                   const float* __restrict__ K,
                                                   float* __restrict__ Rel) {
  __shared__ v2f kPair[2 * N_DIM];                     // 16 KB
  const int bc   = blockIdx.x;                         // b*64 + c
  const int lane = threadIdx.x & 31;
  const int wave = threadIdx.x >> 5;

  const float* kBase = K + (size_t)bc * U_DIM * N_DIM;
  for (int idx = threadIdx.x; idx < 2 * N_DIM; idx += 256) {
    const int half = idx >> 10;
    const int j    = idx & (N_DIM - 1);
    v2f p;
    p.x = kBase[(size_t)(half ? 2 : 0) * N_DIM + j];
    p.y = half ? 0.0f : kBase[(size_t)N_DIM + j];
    kPair[idx] = p;
  }
  __syncthreads();

  const int halfsel = lane >> 4;
  const int lrow    = lane & 15;
  const int i0      = ((blockIdx.y << 3) + wave) << 4; // this wave's i-strip

  const float* qBase = Q + (size_t)bc * U_DIM * N_DIM;
  // A tile (q): lanes 0-15 -> u=0,1 ; lanes 16-31 -> u=2, pad 0
  v2f a;
  a.x = qBase[(size_t)(halfsel ? 2 : 0) * N_DIM + i0 + lrow];
  a.y = halfsel ? 0.0f : qBase[(size_t)N_DIM + i0 + lrow];

  const v2f* kp = kPair + (halfsel << 10) + lrow;

  float rmax[8];
#pragma unroll
  for (int r = 0; r < 8; ++r) rmax[r] = -3.402823466e38f;

#pragma unroll 4
  for (int j0 = 0; j0 < N_DIM; j0 += 16) {
    const v2f bm = kp[j0];                             // one ds_load_b64
    v8f c = {};
    v8f d = __builtin_amdgcn_wmma_f32_16x16x4_f32(false, a, false, bm,
                                                  (short)0, c, false, false);
#pragma unroll
    for (int r = 0; r < 8; ++r) rmax[r] = fmaxf(rmax[r], d[r]);
  }

  // max across the 16 lanes of each half (wave32: offsets 8..1 stay in-half)
#pragma unroll
  for (int off = 8; off >= 1; off >>= 1) {
#pragma unroll
    for (int r = 0; r < 8; ++r)
      rmax[r] = fmaxf(rmax[r], __shfl_xor(rmax[r], off, 32));
  }
  if (lrow == 0) {
    float4* relp = (float4*)(Rel + (size_t)bc * N_DIM + i0 + (halfsel << 3));
    relp[0] = make_float4(0.125f * rmax[0], 0.125f * rmax[1],
                          0.125f * rmax[2], 0.125f * rmax[3]);
    relp[1] = make_float4(0.125f * rmax[4], 0.125f * rmax[5],
                          0.125f * rmax[6], 0.125f * rmax[7]);
  }
}

// ---------------------------------------------------------------------------
// gate+out: g[b,d,i] = sigmoid(sum_c Wg[d,c]*rel[b,c,i] + bg[d])
//           out[b,d,u,i] = x[b,d,u,i] * (1 + g[b,d,i])
// ---------------------------------------------------------------------------
__global__ __launch_bounds__(256) void gate_out(const float* __restrict__ Rel,
                                                const float* __restrict__ Wg,
                                                const float* __restrict__ bg,
                                                const float* __restrict__ X,
                                                float* __restrict__ Out) {
  const int t = blockIdx.x * 256 + threadIdx.x;        // 0 .. B*C*N-1
  if (t >= B_DIM * C_DIM * N_DIM) return;
  const int i = t & (N_DIM - 1);
  const int d = (t >> 10) & (C_DIM - 1);
  const int b = t >> 16;

  float s = bg[d];
  const float* relp = Rel + (size_t)b * C_DIM * N_DIM + i;
  const float* wg   = Wg + d * C_DIM;
#pragma unroll 8
  for (int c = 0; c < C_DIM; ++c) s += wg[c] * relp[(size_t)c * N_DIM];

  const float g = 1.0f / (1.0f + __expf(-s));
  const float m = 1.0f + g;
#pragma unroll
  for (int u = 0; u < U_DIM; ++u) {
    const size_t idx = ((size_t)(b * C_DIM + d) * U_DIM + u) * N_DIM + i;
    Out[idx] = X[idx] * m;
  }
}

// ---------------------------------------------------------------------------
extern "C" void kernel_launch(void* const* d_in, const int* in_sizes, int n_in,
                              void* d_out, int out_size, void* d_ws, size_t ws_size,
                              hipStream_t stream) {
  const float* x  = (const float*)d_in[0];
  const float* y  = (const float*)d_in[1];
  const float* Wq = (const float*)d_in[2];
  const float* Wk = (const float*)d_in[3];
  const float* Wg = (const float*)d_in[4];
  const float* bg = (const float*)d_in[5];
  float* out = (float*)d_out;

  const size_t projElems = (size_t)B_DIM * C_DIM * U_DIM * N_DIM; // 393216
  float* q   = (float*)d_ws;
  float* k   = q + projElems;
  float* rel = k + projElems;                                     // B*C*N

  proj_wmma<<<192, 256, 0, stream>>>(Wq, x, q);
  proj_wmma<<<192, 256, 0, stream>>>(Wk, y, k);
  relmax_wmma<<<dim3(B_DIM * C_DIM, 8), 256, 0, stream>>>(q, k, rel);

  const int total = B_DIM * C_DIM * N_DIM;
  gate_out<<<(total + 255) / 256, 256, 0, stream>>>(rel, Wg, bg, x, out);
}